// VectorQuantizer_v2_27152783245577
// MI455X (gfx1250) — compile-verified
//
#include <hip/hip_runtime.h>
#include <hip/hip_bf16.h>

typedef __attribute__((ext_vector_type(2))) float v2f;
typedef __attribute__((ext_vector_type(8))) float v8f;
typedef __attribute__((ext_vector_type(4))) int   v4i;

#define EMB_DIM    256
#define NCODES     1024
#define SPATIAL    8192                         // 8*32*32
#define NBATCH     4
#define NROWS      (NBATCH * SPATIAL)           // 32768
#define ZQ_SIZE    (NBATCH * EMB_DIM * SPATIAL) // 8388608
#define COMMIT_OFF ZQ_SIZE
#define IDX_OFF    (ZQ_SIZE + 1)
#define CHUNK      32                           // codes per LDS stage (double buffered)
#define NCHUNKS    (NCODES / CHUNK)             // 32
#define LDS_STRIDE 260                          // 256 + 4 pad -> conflict-free ds_load_b64
#define COMMIT_SCALE (0.25f / 8388608.0f)

// ---------------------------------------------------------------------------
// Async global->LDS copy (CDNA5 GLOBAL_LOAD_ASYNC_TO_LDS_B128, ASYNCcnt),
// with a synchronous float4 fallback if the builtin is unavailable.
// ---------------------------------------------------------------------------
#if __has_builtin(__builtin_amdgcn_global_load_async_to_lds_b128)
#define HAVE_ASYNC_LDS 1
#else
#define HAVE_ASYNC_LDS 0
#endif

#if HAVE_ASYNC_LDS
typedef __attribute__((address_space(1))) v4i* gv4i_ptr;   // global int4*
typedef __attribute__((address_space(3))) v4i* lv4i_ptr;   // LDS int4*
#endif

__device__ __forceinline__ void copy16_to_lds(const float* __restrict__ g, float* l)
{
#if HAVE_ASYNC_LDS
    // generic global pointer is numerically the as(1) address; generic LDS
    // pointer carries the LDS byte offset in its low 32 bits (aperture rule).
    __builtin_amdgcn_global_load_async_to_lds_b128(
        (gv4i_ptr)(uintptr_t)g,
        (lv4i_ptr)(uint32_t)(uintptr_t)l,
        0, 0);
#else
    *(float4*)l = *(const float4*)g;
#endif
}

__device__ __forceinline__ void wait_stage()
{
#if HAVE_ASYNC_LDS
#if __has_builtin(__builtin_amdgcn_s_wait_asynccnt)
    __builtin_amdgcn_s_wait_asynccnt(0);
#else
    asm volatile("s_wait_asynccnt 0x0" ::: "memory");
#endif
#endif
}

// Stage one CHUNK x 256 codebook slab into an LDS buffer (rows padded).
__device__ __forceinline__ void stage_chunk(float* dst, const float* __restrict__ cb,
                                            int chunk, int tid)
{
#pragma unroll
    for (int i = tid; i < CHUNK * (EMB_DIM / 4); i += 256) {
        const int r  = i >> 6;                 // 64 float4 per code row
        const int c4 = i & 63;
        copy16_to_lds(cb + (size_t)(chunk * CHUNK + r) * EMB_DIM + 4 * c4,
                      dst + r * LDS_STRIDE + 4 * c4);
    }
}

// ---------------------------------------------------------------------------
// Kernel 1: e2[k] = ||codebook[k]||^2 ; also zero the commit accumulator.
// ---------------------------------------------------------------------------
__global__ void vq_e2(const float* __restrict__ cb, float* __restrict__ e2,
                      float* __restrict__ commit)
{
    const int lane = threadIdx.x & 31;
    const int code = blockIdx.x * (blockDim.x >> 5) + (threadIdx.x >> 5);
    if (blockIdx.x == 0 && threadIdx.x == 0) commit[0] = 0.0f;
    const float* row = cb + (size_t)code * EMB_DIM;
    float s = 0.0f;
#pragma unroll
    for (int i = lane; i < EMB_DIM; i += 32) {
        const float v = row[i];
        s = fmaf(v, v, s);
    }
#pragma unroll
    for (int m = 16; m; m >>= 1) s += __shfl_xor(s, m, 32);
    if (lane == 0) e2[code] = s;
}

// ---------------------------------------------------------------------------
// Kernel 2: argmin_k ( e2[k] - 2 x.e_k ) via V_WMMA_F32_16X16X4_F32.
// 8 waves / block, one 16-row M-tile per wave, full 16x256 A-panel in VGPRs.
// Two independent 16-wide N-tiles per chunk -> two independent WMMA
// accumulator chains (no back-to-back RAW delays). Codebook streamed through
// a double-buffered LDS slab with async copies overlapping the WMMA stream.
// ---------------------------------------------------------------------------
__global__ __launch_bounds__(256, 1)
void vq_argmin(const float* __restrict__ z, const float* __restrict__ cb,
               const float* __restrict__ e2, int* __restrict__ idx_ws,
               float* __restrict__ idx_out)
{
    __shared__ float lds[2][CHUNK * LDS_STRIDE];   // 2 x 33.3 KB

    const int tid  = threadIdx.x;
    const int wave = tid >> 5;
    const int lane = tid & 31;
    const int half = lane >> 4;          // 0 -> K pair {0,1}; 1 -> K pair {2,3}
    const int myN  = lane & 15;
    const int m0   = (blockIdx.x * 8 + wave) * 16;

    // x[row][d] lives at z[b*EMB_DIM*SPATIAL + d*SPATIAL + sp], row = b*SPATIAL+sp
    const int mrow = m0 + myN;
    const float* xrow = z + (size_t)(mrow >> 13) * (EMB_DIM * SPATIAL)
                          + (size_t)(mrow & (SPATIAL - 1));

    // Kick off the first codebook slab before touching z.
    stage_chunk(lds[0], cb, 0, tid);

    // A panel: 128 floats per lane = full K for this lane's row/half.
    float a[128];
#pragma unroll
    for (int t = 0; t < 64; ++t) {
        const int k = 4 * t + 2 * half;
        a[2*t]   = xrow[(size_t)k       * SPATIAL];
        a[2*t+1] = xrow[(size_t)(k + 1) * SPATIAL];
    }

    float bestVal[8];
    int   bestIdx[8];
#pragma unroll
    for (int r = 0; r < 8; ++r) { bestVal[r] = 3.4e38f; bestIdx[r] = 0; }

    wait_stage();
    __syncthreads();

    for (int chunk = 0; chunk < NCHUNKS; ++chunk) {
        const int cur = chunk & 1;
        if (chunk + 1 < NCHUNKS)
            stage_chunk(lds[cur ^ 1], cb, chunk + 1, tid);   // overlaps compute

        // Two independent accumulator chains (N-tiles 0 and 1 of this chunk).
        v8f c0 = {};
        v8f c1 = {};
        const float* bbase0 = &lds[cur][myN * LDS_STRIDE + 2 * half];
        const float* bbase1 = bbase0 + 16 * LDS_STRIDE;
#pragma unroll
        for (int t = 0; t < 64; ++t) {
            v2f av; av.x = a[2*t]; av.y = a[2*t+1];
            const v2f bv0 = *(const v2f*)(bbase0 + 4 * t);   // ds_load_b64
            const v2f bv1 = *(const v2f*)(bbase1 + 4 * t);
            c0 = __builtin_amdgcn_wmma_f32_16x16x4_f32(
                     false, av, false, bv0, (short)0, c0, false, false);
            c1 = __builtin_amdgcn_wmma_f32_16x16x4_f32(
                     false, av, false, bv1, (short)0, c1, false, false);
        }

        const int   n0   = chunk * CHUNK + myN;
        const int   n1   = n0 + 16;
        const float e2v0 = e2[n0];
        const float e2v1 = e2[n1];
#pragma unroll
        for (int r = 0; r < 8; ++r) {
            const float s0 = fmaf(-2.0f, c0[r], e2v0);       // e2 - 2*x.e
            if (s0 < bestVal[r]) { bestVal[r] = s0; bestIdx[r] = n0; }
            const float s1 = fmaf(-2.0f, c1[r], e2v1);
            if (s1 < bestVal[r]) { bestVal[r] = s1; bestIdx[r] = n1; }
        }

        wait_stage();          // next slab fully in LDS
        __syncthreads();       // everyone done reading lds[cur]
    }

    // Cross-lane argmin within each 16-lane half (ties -> smaller index,
    // matching jnp.argmin first-occurrence semantics).
#pragma unroll
    for (int r = 0; r < 8; ++r) {
        float v  = bestVal[r];
        int   ix = bestIdx[r];
#pragma unroll
        for (int m = 1; m < 16; m <<= 1) {
            const float ov = __shfl_xor(v,  m, 32);
            const int   oi = __shfl_xor(ix, m, 32);
            if (ov < v || (ov == v && oi < ix)) { v = ov; ix = oi; }
        }
        bestVal[r] = v; bestIdx[r] = ix;
    }
    if (myN == 0) {
#pragma unroll
        for (int r = 0; r < 8; ++r) {
            const int row = m0 + half * 8 + r;   // C layout: lanes16-31 hold M=r+8
            idx_ws[row]  = bestIdx[r];
            idx_out[row] = (float)bestIdx[r];
        }
    }
}

// ---------------------------------------------------------------------------
// Kernel 3: zq = codebook[idx] scattered back to (b,c,f,h,w); commit loss.
// Threads map to spatial positions -> coalesced z reads / zq writes.
// ---------------------------------------------------------------------------
__global__ void vq_gather(const float* __restrict__ z, const float* __restrict__ cb,
                          const int* __restrict__ idx_ws, float* __restrict__ out)
{
    const int sp  = blockIdx.x * blockDim.x + threadIdx.x;
    const int b   = blockIdx.y;
    const int row = b * SPATIAL + sp;
    const int code = idx_ws[row];

    const float* crow = cb + (size_t)code * EMB_DIM;
    const float* zb   = z   + (size_t)b * (EMB_DIM * SPATIAL) + sp;
    float*       ob   = out + (size_t)b * (EMB_DIM * SPATIAL) + sp;

    float acc = 0.0f;
#pragma unroll 4
    for (int c = 0; c < EMB_DIM; ++c) {
        const float q  = crow[c];                  // L2-resident gather
        const float xv = zb[(size_t)c * SPATIAL];  // coalesced across lanes
        ob[(size_t)c * SPATIAL] = q;               // coalesced store
        const float d = q - xv;
        acc = fmaf(d, d, acc);
    }

#pragma unroll
    for (int m = 16; m; m >>= 1) acc += __shfl_xor(acc, m, 32);

    __shared__ float part[8];
    if ((threadIdx.x & 31) == 0) part[threadIdx.x >> 5] = acc;
    __syncthreads();
    if (threadIdx.x == 0) {
        float tot = 0.0f;
#pragma unroll
        for (int i = 0; i < 8; ++i) tot += part[i];
        atomicAdd(out + COMMIT_OFF, tot * COMMIT_SCALE);
    }
}

// ---------------------------------------------------------------------------
extern "C" void kernel_launch(void* const* d_in, const int* in_sizes, int n_in,
                              void* d_out, int out_size, void* d_ws, size_t ws_size,
                              hipStream_t stream)
{
    const float* z  = (const float*)d_in[0];   // [4,256,8,32,32]
    const float* cb = (const float*)d_in[1];   // [1024,256]
    float* out = (float*)d_out;                // [zq | commit | indices(as float)]

    float* e2_ws  = (float*)d_ws;                        // 1024 floats
    int*   idx_ws = (int*)((char*)d_ws + 4096);          // 32768 ints

    vq_e2<<<NCODES / 8, 256, 0, stream>>>(cb, e2_ws, out + COMMIT_OFF);

    // 32768 rows / 16 per wave / 8 waves per block = 256 blocks
    vq_argmin<<<NROWS / 128, 256, 0, stream>>>(z, cb, e2_ws, idx_ws, out + IDX_OFF);

    vq_gather<<<dim3(SPATIAL / 256, NBATCH), 256, 0, stream>>>(z, cb, idx_ws, out);
}